// DGCNN_semseg_s3dis_9543417331892
// MI455X (gfx1250) — compile-verified
//
#include <hip/hip_runtime.h>
#include <math.h>

typedef __attribute__((ext_vector_type(2))) float v2f;
typedef __attribute__((ext_vector_type(8))) float v8f;

#define NEG_BIG -3.402823466e38f

// ---------------------------------------------------------------------------
// CDNA5 async global->LDS copy (GLOBAL_LOAD_ASYNC_TO_LDS_B32, ASYNCcnt).
// Generic pointers into LDS carry the LDS byte offset in their low 32 bits
// (flat LDS aperture), so truncation yields the
// LDS-address VGPR operand.
// ---------------------------------------------------------------------------
__device__ inline void lds_async_copy_f32(float* dst_lds, const float* src_glb) {
  unsigned dst = (unsigned)(unsigned long long)dst_lds;
  asm volatile("global_load_async_to_lds_b32 %0, %1, off"
               :
               : "v"(dst), "v"(src_glb)
               : "memory");
}
__device__ inline void wait_async0() {
  asm volatile("s_wait_asynccnt 0" ::: "memory");
}

// ---------------------------------------------------------------------------
// WMMA helpers (V_WMMA_F32_16X16X4_F32, wave32)
// D = A(16x4) * B(4x16) + C(16x16), fp32 throughout.
// ---------------------------------------------------------------------------
__device__ inline v8f wmma4(v2f a, v2f b, v8f c) {
  // 8 args: (neg_a, A, neg_b, B, c_mod, C, reuse_a, reuse_b)
  return __builtin_amdgcn_wmma_f32_16x16x4_f32(false, a, false, b, (short)0, c,
                                               false, false);
}

// A-matrix 16x4 fragment from row-major W[ld] at (m0, k0).
// Layout: lanes 0-15 M=0..15 ; VGPR0 = K {0 | 2}, VGPR1 = K {1 | 3} per lane half.
__device__ inline v2f afrag(const float* __restrict__ W, int ld, int m0, int k0) {
  int lane = threadIdx.x & 31;
  int m = m0 + (lane & 15);
  int kb = k0 + ((lane >> 4) << 1);
  v2f a;
  a[0] = W[(size_t)m * ld + kb];
  a[1] = W[(size_t)m * ld + kb + 1];
  return a;
}

// B-matrix 4x16 fragment from an LDS buffer stored [C][16] (stride 16).
__device__ inline v2f bfrag16(const float* __restrict__ F, int k0) {
  int lane = threadIdx.x & 31;
  int n = lane & 15;
  int kb = k0 + ((lane >> 4) << 1);
  v2f b;
  b[0] = F[kb * 16 + n];
  b[1] = F[(kb + 1) * 16 + n];
  return b;
}

// Store a D tile (16x16) into LDS buffer stored [C][16] at row m0.
// Layout: VGPR r: lanes 0-15 -> row r, lanes 16-31 -> row r+8; col = lane&15.
__device__ inline void dstore16(float* __restrict__ H, int m0, v8f d) {
  int lane = threadIdx.x & 31;
  int col = lane & 15;
  int row = m0 + ((lane & 16) ? 8 : 0);
#pragma unroll
  for (int r = 0; r < 8; ++r) H[(row + r) * 16 + col] = d[r];
}

// Inference batchnorm + LeakyReLU(0.2) applied in-register to a D tile whose
// row-0 channel is c0.  bnp = stacked (gamma, beta, mean, var), each length C.
__device__ inline v8f bn_lrelu(v8f d, const float* __restrict__ bnp, int C, int c0) {
  int lane = threadIdx.x & 31;
  int cb = c0 + ((lane & 16) ? 8 : 0);
  v8f o;
#pragma unroll
  for (int r = 0; r < 8; ++r) {
    int ch = cb + r;
    float gg = bnp[ch];
    float bb = bnp[C + ch];
    float mm = bnp[2 * C + ch];
    float vv = bnp[3 * C + ch];
    float s = gg / sqrtf(vv + 1e-5f);
    float y = (d[r] - mm) * s + bb;
    o[r] = (y > 0.f) ? y : 0.2f * y;
  }
  return o;
}

__device__ inline v8f vzero() {
  v8f z;
#pragma unroll
  for (int r = 0; r < 8; ++r) z[r] = 0.f;
  return z;
}

// ---------------------------------------------------------------------------
// 1. transpose (B,C,N) channel-major -> (B,N,C) point-major
// ---------------------------------------------------------------------------
__global__ void transpose_kernel(const float* __restrict__ x,
                                 float* __restrict__ xt, int B, int C, int N) {
  int i = blockIdx.x * blockDim.x + threadIdx.x;
  int total = B * C * N;
  if (i >= total) return;
  int n = i % N;
  int c = (i / N) % C;
  int b = i / (N * C);
  xt[((size_t)b * N + n) * C + c] = x[i];
}

// ---------------------------------------------------------------------------
// 2. fused edge conv: gather(idx) -> [nbr-ctr; ctr] -> conv1(+BN+LReLU)
//    [-> conv2(+BN+LReLU)] -> max over K neighbors.   One wave / 16 points.
//    featT: (B,N,Cin) point-major.  idx==nullptr means neighbor index = k
//    (degenerate dim9 knn on empty features -> stable top-k gives [0..19]).
//    Output: (B,N,64) point-major.
// ---------------------------------------------------------------------------
__global__ __launch_bounds__(32) void edge_conv_kernel(
    const float* __restrict__ featT, const int* __restrict__ idx,
    const float* __restrict__ Wa, const float* __restrict__ bna,
    const float* __restrict__ Wb, const float* __restrict__ bnb,
    float* __restrict__ outT, int N, int Cin, int use2) {
  __shared__ float Ctile[16 * 64];   // center features [point][ch]
  __shared__ float Fbuf[128 * 16];   // edge features   [ch][point]
  __shared__ float Hbuf[64 * 16];    // conv1 output    [ch][point]

  int b = blockIdx.x;
  int p0 = blockIdx.y * 16;
  int lane = threadIdx.x;
  const float* ft = featT + (size_t)b * N * Cin;

  // stage center tile via async data mover
  for (int t = lane; t < 16 * Cin; t += 32) {
    int j = t / Cin, c = t % Cin;
    lds_async_copy_f32(&Ctile[j * 64 + c], &ft[(size_t)(p0 + j) * Cin + c]);
  }
  wait_async0();
  __syncthreads();

  v8f vmax[4];
#pragma unroll
  for (int mt = 0; mt < 4; ++mt)
#pragma unroll
    for (int r = 0; r < 8; ++r) vmax[mt][r] = NEG_BIG;

  int C2 = 2 * Cin;
  int ks1 = C2 >> 2;

  for (int k = 0; k < 20; ++k) {
    // build edge-feature tile for this neighbor slot
    for (int t = lane; t < 16 * Cin; t += 32) {
      int j = t / Cin, c = t % Cin;
      int nj = idx ? idx[((size_t)b * N + p0 + j) * 20 + k] : k;
      float nv = ft[(size_t)nj * Cin + c];
      float cv = Ctile[j * 64 + c];
      Fbuf[c * 16 + j] = nv - cv;
      Fbuf[(Cin + c) * 16 + j] = cv;
    }
    __syncthreads();

    // conv1: 64 x C2
#pragma unroll
    for (int mt = 0; mt < 4; ++mt) {
      v8f acc = vzero();
      for (int ks = 0; ks < ks1; ++ks)
        acc = wmma4(afrag(Wa, C2, mt * 16, ks * 4), bfrag16(Fbuf, ks * 4), acc);
      acc = bn_lrelu(acc, bna, 64, mt * 16);
      if (use2) {
        dstore16(Hbuf, mt * 16, acc);
      } else {
#pragma unroll
        for (int r = 0; r < 8; ++r) vmax[mt][r] = fmaxf(vmax[mt][r], acc[r]);
      }
    }

    if (use2) {
      __syncthreads();
      // conv2: 64 x 64
#pragma unroll
      for (int mt = 0; mt < 4; ++mt) {
        v8f acc = vzero();
#pragma unroll
        for (int ks = 0; ks < 16; ++ks)
          acc = wmma4(afrag(Wb, 64, mt * 16, ks * 4), bfrag16(Hbuf, ks * 4), acc);
        acc = bn_lrelu(acc, bnb, 64, mt * 16);
#pragma unroll
        for (int r = 0; r < 8; ++r) vmax[mt][r] = fmaxf(vmax[mt][r], acc[r]);
      }
    }
    __syncthreads();
  }

  // write max-pooled point features (B,N,64)
  int col = lane & 15;
  int rbase = (lane & 16) ? 8 : 0;
#pragma unroll
  for (int mt = 0; mt < 4; ++mt)
#pragma unroll
    for (int r = 0; r < 8; ++r)
      outT[((size_t)b * N + p0 + col) * 64 + mt * 16 + rbase + r] = vmax[mt][r];
}

// ---------------------------------------------------------------------------
// 3. knn (64-channel, point-major).  One wave per query point.
//    Ranking key s = ||f_m||^2 - 2<f_n,f_m>  (== -pd up to per-row constant),
//    smaller is better; ties -> smaller index (matches stable top_k).
// ---------------------------------------------------------------------------
__global__ __launch_bounds__(256) void knn_kernel(const float* __restrict__ featT,
                                                  int* __restrict__ idx, int N) {
  __shared__ float qs[8][64];
  int wave = threadIdx.x >> 5;
  int lane = threadIdx.x & 31;
  int gq = blockIdx.x * 8 + wave;  // over B*N
  int b = gq / N;
  int n = gq % N;
  const float* ft = featT + (size_t)b * N * 64;

  lds_async_copy_f32(&qs[wave][lane], &ft[(size_t)n * 64 + lane]);
  lds_async_copy_f32(&qs[wave][lane + 32], &ft[(size_t)n * 64 + lane + 32]);
  wait_async0();
  __syncthreads();

  float vals[20];
  int idxs[20];
#pragma unroll
  for (int j = 0; j < 20; ++j) {
    vals[j] = 3.402823466e38f;
    idxs[j] = 0x7fffffff;
  }

  for (int m = lane; m < N; m += 32) {
    const float* fm = ft + (size_t)m * 64;
    float dot = 0.f, nrm = 0.f;
#pragma unroll
    for (int c = 0; c < 64; ++c) {
      float f = fm[c];
      dot += qs[wave][c] * f;
      nrm += f * f;
    }
    float cv = nrm - 2.f * dot;
    int ci = m;
    // bubble-insert into sorted-ascending register list (static indexing)
#pragma unroll
    for (int j = 0; j < 20; ++j) {
      bool better = (cv < vals[j]) || (cv == vals[j] && ci < idxs[j]);
      float tv = vals[j];
      int ti = idxs[j];
      if (better) {
        vals[j] = cv;
        idxs[j] = ci;
        cv = tv;
        ci = ti;
      }
    }
  }

  // 20-round wave merge of 32 sorted lists via butterfly min-reduction
  int* out = idx + (size_t)gq * 20;
  for (int t = 0; t < 20; ++t) {
    float bv = vals[0];
    int bi = idxs[0];
#pragma unroll
    for (int off = 16; off > 0; off >>= 1) {
      float ov = __shfl_xor(bv, off, 32);
      int oi = __shfl_xor(bi, off, 32);
      if (ov < bv || (ov == bv && oi < bi)) {
        bv = ov;
        bi = oi;
      }
    }
    bool won = (vals[0] == bv) && (idxs[0] == bi);
    if (won) {
#pragma unroll
      for (int j = 0; j < 19; ++j) {
        vals[j] = vals[j + 1];
        idxs[j] = idxs[j + 1];
      }
      vals[19] = 3.402823466e38f;
      idxs[19] = 0x7fffffff;
    }
    if (lane == 0) out[t] = bi;
  }
}

// ---------------------------------------------------------------------------
// stage the 192x16 concat tile [x1|x2|x3] into LDS via async data mover
// ---------------------------------------------------------------------------
__device__ inline void stage_concat_tile(float* S, const float* x1t,
                                         const float* x2t, const float* x3t,
                                         int b, int p0, int N) {
  int lane = threadIdx.x & 31;
  for (int t = lane; t < 192 * 16; t += 32) {
    int c = t >> 4, j = t & 15;
    const float* src = (c < 64) ? x1t : (c < 128) ? x2t : x3t;
    int cc = c & 63;
    lds_async_copy_f32(&S[c * 16 + j], &src[((size_t)b * N + p0 + j) * 64 + cc]);
  }
  wait_async0();
}

// ---------------------------------------------------------------------------
// 7. W6 (1024x192) GEMM + BN + LReLU, fused with max over the 16-point tile.
//    Writes per-tile partial maxima (B,1024,NT).
// ---------------------------------------------------------------------------
__global__ __launch_bounds__(32) void gemm6_kernel(
    const float* __restrict__ x1t, const float* __restrict__ x2t,
    const float* __restrict__ x3t, const float* __restrict__ W6,
    const float* __restrict__ bn6, float* __restrict__ partial, int N) {
  __shared__ float S[192 * 16];
  int b = blockIdx.x;
  int tile = blockIdx.y;
  int p0 = tile * 16;
  int lane = threadIdx.x;
  int NT = gridDim.y;

  stage_concat_tile(S, x1t, x2t, x3t, b, p0, N);
  __syncthreads();

  for (int mt = 0; mt < 64; ++mt) {
    v8f acc = vzero();
    for (int ks = 0; ks < 48; ++ks)
      acc = wmma4(afrag(W6, 192, mt * 16, ks * 4), bfrag16(S, ks * 4), acc);
    acc = bn_lrelu(acc, bn6, 1024, mt * 16);
    // max over the 16 columns (points) of this D tile
#pragma unroll
    for (int r = 0; r < 8; ++r) {
      float v = acc[r];
#pragma unroll
      for (int off = 1; off < 16; off <<= 1) v = fmaxf(v, __shfl_xor(v, off, 32));
      if ((lane & 15) == 0) {
        int ch = mt * 16 + ((lane & 16) ? 8 : 0) + r;
        partial[((size_t)b * 1024 + ch) * NT + tile] = v;
      }
    }
  }
}

// 8. reduce partial maxima over tiles -> global feature g (B,1024)
__global__ void reduce_max_kernel(const float* __restrict__ partial,
                                  float* __restrict__ g, int NT, int total) {
  int i = blockIdx.x * blockDim.x + threadIdx.x;
  if (i >= total) return;
  const float* p = partial + (size_t)i * NT;
  float m = NEG_BIG;
  for (int t = 0; t < NT; ++t) m = fmaxf(m, p[t]);
  g[i] = m;
}

// ---------------------------------------------------------------------------
// 9. fused head: W7(512x1216)+BN+LReLU -> W8(256x512)+BN+LReLU -> W9(13x256).
//    Input channels: [g(1024, uniform) | x1(64) | x2(64) | x3(64)].
//    H7/H8 tiles live entirely in LDS.  Output logits (B,13,N).
// ---------------------------------------------------------------------------
__global__ __launch_bounds__(32) void head_kernel(
    const float* __restrict__ g, const float* __restrict__ x1t,
    const float* __restrict__ x2t, const float* __restrict__ x3t,
    const float* __restrict__ W7, const float* __restrict__ bn7,
    const float* __restrict__ W8, const float* __restrict__ bn8,
    const float* __restrict__ W9, float* __restrict__ out, int N) {
  __shared__ float S[192 * 16];
  __shared__ float H7[512 * 16];
  __shared__ float H8[256 * 16];
  int b = blockIdx.x;
  int p0 = blockIdx.y * 16;
  int lane = threadIdx.x;
  const float* gb = g + (size_t)b * 1024;

  stage_concat_tile(S, x1t, x2t, x3t, b, p0, N);
  __syncthreads();

  // pre-touch downstream weights while GEMM7 runs (global_prefetch_b8)
  __builtin_prefetch(W8 + lane * 256, 0, 3);
  __builtin_prefetch(W9 + lane * 64, 0, 3);

  int kbofs = ((lane >> 4) << 1);

  // GEMM7: 512 x 1216
  for (int mt = 0; mt < 32; ++mt) {
    v8f acc = vzero();
    for (int ks = 0; ks < 256; ++ks) {  // global-feature part (uniform per batch)
      int kb = ks * 4 + kbofs;
      v2f bf;
      bf[0] = gb[kb];
      bf[1] = gb[kb + 1];
      acc = wmma4(afrag(W7, 1216, mt * 16, ks * 4), bf, acc);
    }
    for (int ks = 0; ks < 48; ++ks)  // per-point part
      acc = wmma4(afrag(W7, 1216, mt * 16, 1024 + ks * 4), bfrag16(S, ks * 4), acc);
    acc = bn_lrelu(acc, bn7, 512, mt * 16);
    dstore16(H7, mt * 16, acc);
  }
  __syncthreads();

  // GEMM8: 256 x 512
  for (int mt = 0; mt < 16; ++mt) {
    v8f acc = vzero();
    for (int ks = 0; ks < 128; ++ks)
      acc = wmma4(afrag(W8, 512, mt * 16, ks * 4), bfrag16(H7, ks * 4), acc);
    acc = bn_lrelu(acc, bn8, 256, mt * 16);
    dstore16(H8, mt * 16, acc);
  }
  __syncthreads();

  // GEMM9: 13 x 256 (rows >= 13 padded with zero A)
  {
    v8f acc = vzero();
    int m = lane & 15;
    for (int ks = 0; ks < 64; ++ks) {
      int kb = ks * 4 + kbofs;
      v2f a;
      a[0] = (m < 13) ? W9[m * 256 + kb] : 0.f;
      a[1] = (m < 13) ? W9[m * 256 + kb + 1] : 0.f;
      acc = wmma4(a, bfrag16(H8, ks * 4), acc);
    }
    int col = lane & 15;
    int rbase = (lane & 16) ? 8 : 0;
#pragma unroll
    for (int r = 0; r < 8; ++r) {
      int row = rbase + r;
      if (row < 13) out[((size_t)b * 13 + row) * N + p0 + col] = acc[r];
    }
  }
}

// ---------------------------------------------------------------------------
// launcher
// ---------------------------------------------------------------------------
extern "C" void kernel_launch(void* const* d_in, const int* in_sizes, int n_in,
                              void* d_out, int out_size, void* d_ws, size_t ws_size,
                              hipStream_t stream) {
  const int B = 4, Cx = 6;
  const int N = in_sizes[0] / (B * Cx);  // 4096
  const int NT = N / 16;

  const float* x = (const float*)d_in[0];
  const float* W1 = (const float*)d_in[1];
  const float* W2 = (const float*)d_in[2];
  const float* W3 = (const float*)d_in[3];
  const float* W4 = (const float*)d_in[4];
  const float* W5 = (const float*)d_in[5];
  const float* W6 = (const float*)d_in[6];
  const float* W7 = (const float*)d_in[7];
  const float* W8 = (const float*)d_in[8];
  const float* W9 = (const float*)d_in[9];
  const float* bn1 = (const float*)d_in[10];
  const float* bn2 = (const float*)d_in[11];
  const float* bn3 = (const float*)d_in[12];
  const float* bn4 = (const float*)d_in[13];
  const float* bn5 = (const float*)d_in[14];
  const float* bn6 = (const float*)d_in[15];
  const float* bn7 = (const float*)d_in[16];
  const float* bn8 = (const float*)d_in[17];
  float* out = (float*)d_out;

  // workspace layout (floats)
  float* ws = (float*)d_ws;
  float* xt = ws;                       // B*N*6
  float* x1t = xt + (size_t)B * N * 6;  // B*N*64
  float* x2t = x1t + (size_t)B * N * 64;
  float* x3t = x2t + (size_t)B * N * 64;
  float* partial = x3t + (size_t)B * N * 64;  // B*1024*NT
  float* gfeat = partial + (size_t)B * 1024 * NT;  // B*1024
  int* idx1 = (int*)(gfeat + (size_t)B * 1024);    // B*N*20
  int* idx2 = idx1 + (size_t)B * N * 20;

  // 1. transpose x -> (B,N,6)
  {
    int total = B * Cx * N;
    transpose_kernel<<<(total + 255) / 256, 256, 0, stream>>>(x, xt, B, Cx, N);
  }
  dim3 tiles(B, N / 16);

  // 2. edge conv block 1 (degenerate knn: neighbor index = k)
  edge_conv_kernel<<<tiles, 32, 0, stream>>>(xt, nullptr, W1, bn1, W2, bn2, x1t,
                                             N, 6, 1);
  // 3. knn on x1
  knn_kernel<<<(B * N) / 8, 256, 0, stream>>>(x1t, idx1, N);
  // 4. edge conv block 2
  edge_conv_kernel<<<tiles, 32, 0, stream>>>(x1t, idx1, W3, bn3, W4, bn4, x2t,
                                             N, 64, 1);
  // 5. knn on x2
  knn_kernel<<<(B * N) / 8, 256, 0, stream>>>(x2t, idx2, N);
  // 6. edge conv block 3 (single conv)
  edge_conv_kernel<<<tiles, 32, 0, stream>>>(x2t, idx2, W5, bn5, nullptr, nullptr,
                                             x3t, N, 64, 0);
  // 7. W6 GEMM + fused tile max
  gemm6_kernel<<<tiles, 32, 0, stream>>>(x1t, x2t, x3t, W6, bn6, partial, N);
  // 8. reduce to global feature
  {
    int total = B * 1024;
    reduce_max_kernel<<<(total + 255) / 256, 256, 0, stream>>>(partial, gfeat,
                                                               NT, total);
  }
  // 9. fused head -> logits (B,13,N)
  head_kernel<<<tiles, 32, 0, stream>>>(gfeat, x1t, x2t, x3t, W7, bn7, W8, bn8,
                                        W9, out, N);
}